// SCD_GraphLayer_71700184039608
// MI455X (gfx1250) — compile-verified
//
#include <hip/hip_runtime.h>
#include <hip/hip_bf16.h>

// ---------------------------------------------------------------------------
// SCD GAT layer for MI455X (gfx1250, wave32).
//  - z = h @ W^T via V_WMMA_F32_16X16X4_F32 (full fp32 tensor-core GEMM)
//  - edge softmax via per-node scalar scores + L2-resident atomics
// ---------------------------------------------------------------------------

typedef __attribute__((ext_vector_type(2))) float v2f;
typedef __attribute__((ext_vector_type(8))) float v8f;

// order-preserving float<->uint encoding for atomic float-max
__device__ __forceinline__ unsigned encf(float f) {
    unsigned b = __float_as_uint(f);
    return (b & 0x80000000u) ? ~b : (b | 0x80000000u);
}
__device__ __forceinline__ float decf(unsigned u) {
    unsigned b = (u & 0x80000000u) ? (u & 0x7fffffffu) : ~u;
    return __uint_as_float(b);
}

__device__ __forceinline__ void atomAddF(float* p, float v) {
    __hip_atomic_fetch_add(p, v, __ATOMIC_RELAXED, __HIP_MEMORY_SCOPE_AGENT);
}
__device__ __forceinline__ void atomMaxU(unsigned* p, unsigned v) {
    __hip_atomic_fetch_max(p, v, __ATOMIC_RELAXED, __HIP_MEMORY_SCOPE_AGENT);
}

__device__ __forceinline__ float dot4(float4 a, float4 b) {
    return a.x * b.x + a.y * b.y + a.z * b.z + a.w * b.w;
}

// ---------------------------------------------------------------------------
// K1: Z[n,128] = H[n,128] @ W^T, H row i = (i < nA ? embA[i] : embB[i-nA]).
// One block = 16 rows; 8 waves each own a 16-col tile; A tile staged in LDS.
// Per wave: 32 x v_wmma_f32_16x16x4_f32 (K stepped by 4).
// ---------------------------------------------------------------------------
__global__ __launch_bounds__(256)
void gat_fc_wmma(const float* __restrict__ embA, int nA,
                 const float* __restrict__ embB,
                 const float* __restrict__ W,
                 float* __restrict__ Z, int nTotal)
{
    __shared__ float sH[16][132];          // 132-float row pitch: conflict-free
    const int row0 = blockIdx.x * 16;
    const int t = threadIdx.x;

    // stage 16x128 A-tile: each thread copies 8 consecutive floats
    {
        int r = t >> 4;                    // 0..15
        int c = (t & 15) * 8;              // 0..120
        int gr = row0 + r;
        if (gr >= nTotal) gr = nTotal - 1; // clamp (sizes are exact multiples)
        const float* src = (gr < nA) ? (embA + (size_t)gr * 128)
                                     : (embB + (size_t)(gr - nA) * 128);
        *(float4*)(&sH[r][c])     = *(const float4*)(src + c);
        *(float4*)(&sH[r][c + 4]) = *(const float4*)(src + c + 4);
    }
    __syncthreads();

    const int lane = t & 31;
    const int colo = (t >> 5) * 16;        // wave -> 16-col tile of Z
    const int mn   = lane & 15;            // A: M index / B: N index
    const int kh   = (lane >> 4) * 2;      // K sub-offset within 4-wide step

    const float* arow = &sH[mn][kh];                         // A[m, kb+kh..+1]
    const float* wrow = W + (size_t)(colo + mn) * 128 + kh;  // B[k,n]=W[n,k]

    v8f acc = {};
#pragma unroll
    for (int kb = 0; kb < 128; kb += 4) {
        v2f a = *(const v2f*)(arow + kb);
        v2f b = *(const v2f*)(wrow + kb);
        acc = __builtin_amdgcn_wmma_f32_16x16x4_f32(
            false, a, false, b, (short)0, acc, false, false);
    }

    // D layout: VGPR r holds M = r + 8*(lane>=16), N = lane&15
    const int mb = (lane >> 4) * 8;
    float* zout = Z + (size_t)(row0 + mb) * 128 + colo + mn;
#pragma unroll
    for (int r = 0; r < 8; ++r)
        if (row0 + mb + r < nTotal) zout[(size_t)r * 128] = acc[r];
}

// ---------------------------------------------------------------------------
// K2: per-node attention scalars es = z . a[:128], ed = z . a[128:]
// one wave per node
// ---------------------------------------------------------------------------
__global__ __launch_bounds__(256)
void node_scores(const float* __restrict__ Z, const float* __restrict__ attnW,
                 float* __restrict__ es, float* __restrict__ ed, int n)
{
    int node = blockIdx.x * 8 + (threadIdx.x >> 5);
    int lane = threadIdx.x & 31;
    if (node >= n) return;
    float4 z4 = *(const float4*)(Z + (size_t)node * 128 + lane * 4);
    float4 al = *(const float4*)(attnW + lane * 4);
    float4 ah = *(const float4*)(attnW + 128 + lane * 4);
    float p0 = dot4(z4, al), p1 = dot4(z4, ah);
#pragma unroll
    for (int off = 16; off; off >>= 1) {
        p0 += __shfl_xor(p0, off, 32);
        p1 += __shfl_xor(p1, off, 32);
    }
    if (lane == 0) { es[node] = p0; ed[node] = p1; }
}

// ---------------------------------------------------------------------------
// K3/K4: segment max and segment sum(exp) over edges (thread per edge)
// ---------------------------------------------------------------------------
__global__ void edge_max(const int* __restrict__ src, const int* __restrict__ dst,
                         const float* __restrict__ es, const float* __restrict__ ed,
                         unsigned* __restrict__ mU, int base, int nE)
{
    int i = blockIdx.x * blockDim.x + threadIdx.x;
    if (i >= nE) return;
    float e = es[src[i]] + ed[dst[i]];
    atomMaxU(&mU[dst[i] - base], encf(e));
}

__global__ void edge_denom(const int* __restrict__ src, const int* __restrict__ dst,
                           const float* __restrict__ es, const float* __restrict__ ed,
                           const unsigned* __restrict__ mU, float* __restrict__ denom,
                           int base, int nE)
{
    int i = blockIdx.x * blockDim.x + threadIdx.x;
    if (i >= nE) return;
    int d = dst[i] - base;
    float ex = __expf(es[src[i]] + ed[dst[i]] - decf(mU[d]));
    atomAddF(&denom[d], ex);
}

// ---------------------------------------------------------------------------
// K5: weighted scatter acc[dst-base] += alpha * z[src]   (wave per edge)
// ---------------------------------------------------------------------------
__global__ __launch_bounds__(256)
void edge_scatter(const int* __restrict__ src, const int* __restrict__ dst,
                  const float* __restrict__ es, const float* __restrict__ ed,
                  const unsigned* __restrict__ mU, const float* __restrict__ denom,
                  const float* __restrict__ Z, float* __restrict__ acc,
                  int base, int nE)
{
    int e = blockIdx.x * 8 + (threadIdx.x >> 5);
    int lane = threadIdx.x & 31;
    if (e >= nE) return;
    int s = src[e], d = dst[e];
    int dl = d - base;
    float alpha = __expf(es[s] + ed[d] - decf(mU[dl])) / denom[dl];
    float4 z4 = *(const float4*)(Z + (size_t)s * 128 + lane * 4);
    float* out = acc + (size_t)dl * 128 + lane * 4;
    atomAddF(out + 0, alpha * z4.x);
    atomAddF(out + 1, alpha * z4.y);
    atomAddF(out + 2, alpha * z4.z);
    atomAddF(out + 3, alpha * z4.w);
}

// ---------------------------------------------------------------------------
// Epilogues
// ---------------------------------------------------------------------------
__global__ void final_add(const float* __restrict__ a, const float* __restrict__ b,
                          float* __restrict__ out, int n)
{
    int i = blockIdx.x * blockDim.x + threadIdx.x;
    if (i < n) out[i] = a[i] + b[i];      // softmax over a single logit == 1
}

__global__ __launch_bounds__(256)
void final_exer(const float* __restrict__ exer,
                const float* __restrict__ accEfs, const float* __restrict__ accEfk,
                const float* __restrict__ W0, const float* __restrict__ b0,
                const float* __restrict__ W1, const float* __restrict__ b1,
                float* __restrict__ out, int E)
{
    int i = blockIdx.x * 8 + (threadIdx.x >> 5);
    int lane = threadIdx.x & 31;
    if (i >= E) return;
    size_t o = (size_t)i * 128 + lane * 4;
    float4 x = *(const float4*)(exer + o);
    float4 f = *(const float4*)(accEfs + o);
    float4 k = *(const float4*)(accEfk + o);
    float4 w0a = *(const float4*)(W0 + lane * 4);
    float4 w0b = *(const float4*)(W0 + 128 + lane * 4);
    float4 w1a = *(const float4*)(W1 + lane * 4);
    float4 w1b = *(const float4*)(W1 + 128 + lane * 4);
    float p0 = dot4(x, w0a) + dot4(f, w0b);
    float p1 = dot4(x, w1a) + dot4(k, w1b);
#pragma unroll
    for (int off = 16; off; off >>= 1) {
        p0 += __shfl_xor(p0, off, 32);
        p1 += __shfl_xor(p1, off, 32);
    }
    float sc0 = p0 + b0[0], sc1 = p1 + b1[0];
    float mm = fmaxf(sc0, sc1);
    float e0 = __expf(sc0 - mm), e1 = __expf(sc1 - mm);
    float inv = 1.0f / (e0 + e1);
    float s0 = e0 * inv, s1 = e1 * inv;
    float4 r;
    r.x = x.x + s0 * f.x + s1 * k.x;
    r.y = x.y + s0 * f.y + s1 * k.y;
    r.z = x.z + s0 * f.z + s1 * k.z;
    r.w = x.w + s0 * f.w + s1 * k.w;
    *(float4*)(out + o) = r;
}

// ---------------------------------------------------------------------------
extern "C" void kernel_launch(void* const* d_in, const int* in_sizes, int n_in,
                              void* d_out, int out_size, void* d_ws, size_t ws_size,
                              hipStream_t stream) {
    const float* stu  = (const float*)d_in[0];
    const float* exer = (const float*)d_in[1];
    const float* kn   = (const float*)d_in[2];
    const float* sfe_fcW = (const float*)d_in[3];  const float* sfe_aW = (const float*)d_in[4];
    const float* efs_fcW = (const float*)d_in[5];  const float* efs_aW = (const float*)d_in[6];
    const float* efk_fcW = (const float*)d_in[7];  const float* efk_aW = (const float*)d_in[8];
    const float* kfe_fcW = (const float*)d_in[9];  const float* kfe_aW = (const float*)d_in[10];
    const float* e_a0W = (const float*)d_in[13]; const float* e_a0b = (const float*)d_in[14];
    const float* e_a1W = (const float*)d_in[15]; const float* e_a1b = (const float*)d_in[16];
    const int* sfe_src = (const int*)d_in[19]; const int* sfe_dst = (const int*)d_in[20];
    const int* efs_src = (const int*)d_in[21]; const int* efs_dst = (const int*)d_in[22];
    const int* efk_src = (const int*)d_in[23]; const int* efk_dst = (const int*)d_in[24];
    const int* kfe_src = (const int*)d_in[25]; const int* kfe_dst = (const int*)d_in[26];
    float* out = (float*)d_out;

    const int KD  = 128;
    const int S   = in_sizes[0] / KD;      // 40000
    const int E   = in_sizes[1] / KD;      // 18000
    const int KN  = in_sizes[2] / KD;      // 128
    const int nSE = S + E;                 // 58000 node space for s_e graphs
    const int nEK = E + KN;                // 18128 node space for e_k graphs
    const int nse = in_sizes[19];          // 1,500,000
    const int nek = in_sizes[23];          // 180,000

    // ---- workspace carve-up (floats) ----
    float* p = (float*)d_ws;
    float* z_sfe = p; p += (size_t)nSE * KD;
    float* z_efs = p; p += (size_t)nSE * KD;
    float* z_efk = p; p += (size_t)nEK * KD;
    float* z_kfe = p; p += (size_t)nEK * KD;
    float* es_sfe = p; p += nSE; float* ed_sfe = p; p += nSE;
    float* es_efs = p; p += nSE; float* ed_efs = p; p += nSE;
    float* es_efk = p; p += nEK; float* ed_efk = p; p += nEK;
    float* es_kfe = p; p += nEK; float* ed_kfe = p; p += nEK;
    // zero-initialized region (one contiguous memset)
    float* zero0 = p;
    float* acc_sfe = p; p += (size_t)S  * KD;   // s_from_e[:S]
    float* acc_efs = p; p += (size_t)E  * KD;   // e_from_s[S:]
    float* acc_efk = p; p += (size_t)E  * KD;   // e_from_k[:E]
    float* acc_kfe = p; p += (size_t)KN * KD;   // k_from_e[E:]
    unsigned* mU_sfe = (unsigned*)p; p += S;  float* dn_sfe = p; p += S;
    unsigned* mU_efs = (unsigned*)p; p += E;  float* dn_efs = p; p += E;
    unsigned* mU_efk = (unsigned*)p; p += E;  float* dn_efk = p; p += E;
    unsigned* mU_kfe = (unsigned*)p; p += KN; float* dn_kfe = p; p += KN;
    size_t zbytes = (size_t)((char*)p - (char*)zero0);
    hipMemsetAsync(zero0, 0, zbytes, stream);   // mU==0 encodes -inf-like identity

    // ---- 1) z = h @ W^T (fp32 WMMA) ----
    gat_fc_wmma<<<(nSE + 15) / 16, 256, 0, stream>>>(exer, E, stu, sfe_fcW, z_sfe, nSE);
    gat_fc_wmma<<<(nSE + 15) / 16, 256, 0, stream>>>(exer, E, stu, efs_fcW, z_efs, nSE);
    gat_fc_wmma<<<(nEK + 15) / 16, 256, 0, stream>>>(exer, E, kn,  efk_fcW, z_efk, nEK);
    gat_fc_wmma<<<(nEK + 15) / 16, 256, 0, stream>>>(exer, E, kn,  kfe_fcW, z_kfe, nEK);

    // ---- 2) per-node scalar attention scores ----
    node_scores<<<(nSE + 7) / 8, 256, 0, stream>>>(z_sfe, sfe_aW, es_sfe, ed_sfe, nSE);
    node_scores<<<(nSE + 7) / 8, 256, 0, stream>>>(z_efs, efs_aW, es_efs, ed_efs, nSE);
    node_scores<<<(nEK + 7) / 8, 256, 0, stream>>>(z_efk, efk_aW, es_efk, ed_efk, nEK);
    node_scores<<<(nEK + 7) / 8, 256, 0, stream>>>(z_kfe, kfe_aW, es_kfe, ed_kfe, nEK);

    // ---- 3) segment max / 4) segment sum(exp) / 5) weighted scatter ----
    int gse = (nse + 255) / 256, gek = (nek + 255) / 256;
    edge_max<<<gse, 256, 0, stream>>>(sfe_src, sfe_dst, es_sfe, ed_sfe, mU_sfe, 0, nse);
    edge_max<<<gse, 256, 0, stream>>>(efs_src, efs_dst, es_efs, ed_efs, mU_efs, S, nse);
    edge_max<<<gek, 256, 0, stream>>>(efk_src, efk_dst, es_efk, ed_efk, mU_efk, 0, nek);
    edge_max<<<gek, 256, 0, stream>>>(kfe_src, kfe_dst, es_kfe, ed_kfe, mU_kfe, E, nek);

    edge_denom<<<gse, 256, 0, stream>>>(sfe_src, sfe_dst, es_sfe, ed_sfe, mU_sfe, dn_sfe, 0, nse);
    edge_denom<<<gse, 256, 0, stream>>>(efs_src, efs_dst, es_efs, ed_efs, mU_efs, dn_efs, S, nse);
    edge_denom<<<gek, 256, 0, stream>>>(efk_src, efk_dst, es_efk, ed_efk, mU_efk, dn_efk, 0, nek);
    edge_denom<<<gek, 256, 0, stream>>>(kfe_src, kfe_dst, es_kfe, ed_kfe, mU_kfe, dn_kfe, E, nek);

    edge_scatter<<<(nse + 7) / 8, 256, 0, stream>>>(sfe_src, sfe_dst, es_sfe, ed_sfe,
                                                    mU_sfe, dn_sfe, z_sfe, acc_sfe, 0, nse);
    edge_scatter<<<(nse + 7) / 8, 256, 0, stream>>>(efs_src, efs_dst, es_efs, ed_efs,
                                                    mU_efs, dn_efs, z_efs, acc_efs, S, nse);
    edge_scatter<<<(nek + 7) / 8, 256, 0, stream>>>(efk_src, efk_dst, es_efk, ed_efk,
                                                    mU_efk, dn_efk, z_efk, acc_efk, 0, nek);
    edge_scatter<<<(nek + 7) / 8, 256, 0, stream>>>(kfe_src, kfe_dst, es_kfe, ed_kfe,
                                                    mU_kfe, dn_kfe, z_kfe, acc_kfe, E, nek);

    // ---- 6) epilogues: ult_stu | ult_exer | ult_kn (concatenated flat) ----
    final_add<<<((S * KD) + 255) / 256, 256, 0, stream>>>(stu, acc_sfe, out, S * KD);
    final_exer<<<(E + 7) / 8, 256, 0, stream>>>(exer, acc_efs, acc_efk,
                                                e_a0W, e_a0b, e_a1W, e_a1b,
                                                out + (size_t)S * KD, E);
    final_add<<<((KN * KD) + 255) / 256, 256, 0, stream>>>(kn, acc_kfe,
                                                           out + (size_t)(S + E) * KD, KN * KD);
}